// Model_13374528160105
// MI455X (gfx1250) — compile-verified
//
#include <hip/hip_runtime.h>
#include <hip/hip_bf16.h>

#define N_NODES 100000
#define N_HIGH  20000
#define N_EDGES 200000
#define D       128
#define HEADS   4
#define HD      (HEADS * D)   // 512
#define EPS_F   1e-16f

typedef __attribute__((ext_vector_type(2))) float        v2f;
typedef __attribute__((ext_vector_type(8))) float        v8f;
typedef __attribute__((ext_vector_type(4))) unsigned int v4u;
typedef __attribute__((ext_vector_type(8))) int          v8i;
typedef __attribute__((ext_vector_type(4))) int          v4i;

// ---------------- workspace layout (floats) ----------------
static constexpr size_t OFF_SEG   = 0;                                   // N_HIGH*D
static constexpr size_t OFF_CNT   = OFF_SEG   + (size_t)N_HIGH * D;      // N_HIGH
static constexpr size_t OFF_ASUM  = OFF_CNT   + (size_t)N_HIGH;          // N_HIGH*HEADS
static constexpr size_t OFF_TMP   = OFF_ASUM  + (size_t)N_HIGH * HEADS;  // N_HIGH*HD
static constexpr size_t ZERO_FLTS = OFF_TMP   + (size_t)N_HIGH * HD;     // zeroed prefix
static constexpr size_t OFF_SNODE = ZERO_FLTS;                           // N_NODES*HEADS
static constexpr size_t OFF_SHIGH = OFF_SNODE + (size_t)N_NODES * HEADS; // N_HIGH*HEADS
static constexpr size_t OFF_ALPHA = OFF_SHIGH + (size_t)N_HIGH * HEADS;  // N_EDGES*HEADS
static constexpr size_t OFF_AVN   = OFF_ALPHA + (size_t)N_EDGES * HEADS; // D*HEADS
static constexpr size_t OFF_AVH   = OFF_AVN   + (size_t)D * HEADS;       // D*HEADS
static constexpr size_t OFF_AMAX  = OFF_AVH   + (size_t)D * HEADS;       // N_HIGH*HEADS (bits)

// float atomic-max via signed/unsigned int atomics (init word must be 0xFF800000 = -inf)
__device__ __forceinline__ void atomicMaxF32(float* addr, float v) {
    if (v >= 0.0f) atomicMax((int*)addr, __float_as_int(v));
    else           atomicMin((unsigned int*)addr, __float_as_uint(v));
}

// ---------------- K0: init amax to -inf ----------------
__global__ void k_init_amax(unsigned int* amax) {
    int i = blockIdx.x * blockDim.x + threadIdx.x;
    if (i < N_HIGH * HEADS) amax[i] = 0xFF800000u;
}

// ---------------- K1: fold attention vectors through projections ----------------
// avn[din,h] = sum_d W_src[din, h*D+d] * att_dst[h,d]
// avh[din,h] = sum_d W_dst[din, h*D+d] * att_src[h,d]
__global__ void k_avecs(const float* __restrict__ Wsrc, const float* __restrict__ Wdst,
                        const float* __restrict__ att_src, const float* __restrict__ att_dst,
                        float* __restrict__ avn, float* __restrict__ avh) {
    int i = blockIdx.x * blockDim.x + threadIdx.x;
    if (i >= 2 * D * HEADS) return;
    int which = i / (D * HEADS);
    int r   = i % (D * HEADS);
    int din = r / HEADS;
    int h   = r % HEADS;
    const float* W   = which ? Wdst    : Wsrc;
    const float* att = which ? att_src : att_dst;
    float s = 0.0f;
    #pragma unroll 8
    for (int d = 0; d < D; ++d)
        s += W[(size_t)din * HD + h * D + d] * att[h * D + d];
    (which ? avh : avn)[din * HEADS + h] = s;
}

// ---------------- K2: segment-sum of gathered node features (one wave32 / edge) ----------------
__global__ void k_scatter_seg(const float* __restrict__ nf, const int* __restrict__ src,
                              const int* __restrict__ dst, float* __restrict__ seg,
                              float* __restrict__ cnt) {
    int wave = (blockIdx.x * blockDim.x + threadIdx.x) >> 5;
    int lane = threadIdx.x & 31;
    if (wave >= N_EDGES) return;
    int s = src[wave], g = dst[wave];
    const float* row = nf + (size_t)s * D;
    float* out = seg + (size_t)g * D;
    #pragma unroll
    for (int j = 0; j < 4; ++j) {
        int d = lane + 32 * j;
        atomicAdd(&out[d], row[d]);
    }
    if (lane == 0) atomicAdd(&cnt[g], 1.0f);
}

// ---------------- K3: per-node attention scores (one wave32 / node) ----------------
__global__ void k_snode(const float* __restrict__ nf, const float* __restrict__ avn,
                        float* __restrict__ snode) {
    int wave = (blockIdx.x * blockDim.x + threadIdx.x) >> 5;
    int lane = threadIdx.x & 31;
    if (wave >= N_NODES) return;
    const float* row = nf + (size_t)wave * D;
    float a0 = 0.f, a1 = 0.f, a2 = 0.f, a3 = 0.f;
    #pragma unroll
    for (int j = 0; j < 4; ++j) {
        int d = lane + 32 * j;
        float v = row[d];
        const float* a = avn + d * HEADS;
        a0 += v * a[0]; a1 += v * a[1]; a2 += v * a[2]; a3 += v * a[3];
    }
    #pragma unroll
    for (int m = 16; m >= 1; m >>= 1) {
        a0 += __shfl_xor(a0, m, 32);
        a1 += __shfl_xor(a1, m, 32);
        a2 += __shfl_xor(a2, m, 32);
        a3 += __shfl_xor(a3, m, 32);
    }
    if (lane == 0) {
        float* o = snode + (size_t)wave * HEADS;
        o[0] = a0; o[1] = a1; o[2] = a2; o[3] = a3;
    }
}

// ---------------- K4: per-hyperedge scores from mean features (one wave32 / hyperedge) ----------------
__global__ void k_shigh(const float* __restrict__ seg, const float* __restrict__ cnt,
                        const float* __restrict__ avh, float* __restrict__ shigh) {
    int wave = (blockIdx.x * blockDim.x + threadIdx.x) >> 5;
    int lane = threadIdx.x & 31;
    if (wave >= N_HIGH) return;
    float c = cnt[wave];
    float inv = 1.0f / fmaxf(c, 1.0f);
    const float* row = seg + (size_t)wave * D;
    float a0 = 0.f, a1 = 0.f, a2 = 0.f, a3 = 0.f;
    #pragma unroll
    for (int j = 0; j < 4; ++j) {
        int d = lane + 32 * j;
        float v = row[d] * inv;
        const float* a = avh + d * HEADS;
        a0 += v * a[0]; a1 += v * a[1]; a2 += v * a[2]; a3 += v * a[3];
    }
    #pragma unroll
    for (int m = 16; m >= 1; m >>= 1) {
        a0 += __shfl_xor(a0, m, 32);
        a1 += __shfl_xor(a1, m, 32);
        a2 += __shfl_xor(a2, m, 32);
        a3 += __shfl_xor(a3, m, 32);
    }
    if (lane == 0) {
        float* o = shigh + (size_t)wave * HEADS;
        o[0] = a0; o[1] = a1; o[2] = a2; o[3] = a3;
    }
}

// ---------------- K5: logits + leaky-relu + segment max ----------------
__global__ void k_alpha_max(const int* __restrict__ src, const int* __restrict__ dst,
                            const float* __restrict__ snode, const float* __restrict__ shigh,
                            float* __restrict__ alpha, float* __restrict__ amax) {
    int e = blockIdx.x * blockDim.x + threadIdx.x;
    if (e >= N_EDGES) return;
    int s = src[e], g = dst[e];
    #pragma unroll
    for (int h = 0; h < HEADS; ++h) {
        float x = shigh[(size_t)g * HEADS + h] + snode[(size_t)s * HEADS + h];
        x = x > 0.0f ? x : 0.2f * x;
        alpha[(size_t)e * HEADS + h] = x;
        atomicMaxF32(&amax[(size_t)g * HEADS + h], x);
    }
}

// ---------------- K6: exp(x - max) + segment sum ----------------
__global__ void k_alpha_exp(const int* __restrict__ dst, float* __restrict__ alpha,
                            const float* __restrict__ amax, float* __restrict__ asum) {
    int i = blockIdx.x * blockDim.x + threadIdx.x;   // over E*HEADS
    if (i >= N_EDGES * HEADS) return;
    int e = i / HEADS, h = i % HEADS;
    int g = dst[e];
    float ex = __expf(alpha[i] - amax[(size_t)g * HEADS + h]);
    alpha[i] = ex;
    atomicAdd(&asum[(size_t)g * HEADS + h], ex);
}

// ---------------- K7: normalize, emit alpha, scatter 0.25*alpha*nn into tmp ----------------
__global__ void k_norm_scatter(const float* __restrict__ nf, const int* __restrict__ src,
                               const int* __restrict__ dst, const float* __restrict__ alpha_ex,
                               const float* __restrict__ asum, float* __restrict__ alpha_out,
                               float* __restrict__ tmp) {
    int wave = (blockIdx.x * blockDim.x + threadIdx.x) >> 5;
    int lane = threadIdx.x & 31;
    if (wave >= N_EDGES) return;
    int s = src[wave], g = dst[wave];
    float a = 0.0f;
    if (lane < HEADS) {
        a = alpha_ex[(size_t)wave * HEADS + lane] /
            (asum[(size_t)g * HEADS + lane] + EPS_F);
        alpha_out[(size_t)wave * HEADS + lane] = a;
    }
    float a0 = 0.25f * __shfl(a, 0, 32);
    float a1 = 0.25f * __shfl(a, 1, 32);
    float a2 = 0.25f * __shfl(a, 2, 32);
    float a3 = 0.25f * __shfl(a, 3, 32);
    const float* row = nf + (size_t)s * D;
    float* t = tmp + (size_t)g * HD;
    #pragma unroll
    for (int j = 0; j < 4; ++j) {
        int d = lane + 32 * j;
        float v = row[d];
        atomicAdd(&t[0 * D + d], a0 * v);
        atomicAdd(&t[1 * D + d], a1 * v);
        atomicAdd(&t[2 * D + d], a2 * v);
        atomicAdd(&t[3 * D + d], a3 * v);
    }
}

// ---------------- K8: fused WMMA double-GEMM with TDM-staged A tile ----------------
// A tile (16 x 512 of tmp) is DMA'd into LDS by the Tensor Data Mover with
// pad_interval=256 DWORDs / pad_amount=2 DWORDs -> effective row stride 516
// floats (516 % 64 = 4, so the 16 A-rows land in 16 distinct LDS banks).
// H1[g,dout] = sum_{h,din} tmp[g, h*128+din] * W_src[din, h*128+dout]  ([16x512] x [512x128])
// OUT = H1 @ W_high + b_high                                           ([16x128] x [128x128])
// Block = 128 threads (4 wave32s); wave w owns output n-tiles {2w, 2w+1}.
#define T_STRIDE   516   // 512 data + 2x2 pad DWORDs per row
#define LDS_STRIDE 132
__global__ void __launch_bounds__(128)
k_gemm(const float* __restrict__ tmp, const float* __restrict__ Wsrc,
       const float* __restrict__ Whigh, const float* __restrict__ bias,
       float* __restrict__ out) {
    __shared__ float Tlds[16 * T_STRIDE];    // 33,024 B
    __shared__ float h1[16 * LDS_STRIDE];    //  8,448 B
    const int g0   = blockIdx.x * 16;
    const int wave = threadIdx.x >> 5;
    const int lane = threadIdx.x & 31;
    const int half = lane >> 4;          // K-half select (+2) per ISA f32 A/B layout
    const int l15  = lane & 15;

#if defined(__HIP_DEVICE_COMPILE__)
    if (wave == 0) {
        // ---- Tensor DMA descriptor (D#) ----
        const unsigned long long ga = (unsigned long long)(const void*)(tmp + (size_t)g0 * HD);
        // flat LDS address low 32 bits == LDS byte offset (ISA aperture mapping)
        const unsigned lds_off = (unsigned)(unsigned long long)(const void*)&Tlds[0];
        v4u g0d;
        g0d.x = 1u;                                            // count=1, user mode
        g0d.y = lds_off;                                       // lds_addr (bytes)
        g0d.z = (unsigned)(ga & 0xFFFFFFFFull);                // global_addr[31:0]
        g0d.w = (unsigned)((ga >> 32) & 0x1FFFFFFull)          // global_addr[56:32]
                | (2u << 30);                                  // type = 2 (image)
        v8i g1d;
        g1d[0] = (int)((2u << 16)        // data_size = 4 bytes
                     | (1u << 20)        // pad_enable
                     | (7u << 22)        // pad_interval: every 256 DWORDs
                     | (1u << 25));      // pad_amount: 2 DWORDs
        g1d[1] = (int)(512u << 16);      // tensor_dim0 = 512 (bits 79:48 low half)
        g1d[2] = (int)(16u  << 16);      // tensor_dim1 = 16  (bits 111:80 low half)
        g1d[3] = (int)(512u << 16);      // tile_dim0   = 512 (bits 127:112)
        g1d[4] = 16;                     // tile_dim1   = 16, tile_dim2 = 0
        g1d[5] = 512;                    // tensor_dim0_stride = 512
        g1d[6] = 0;
        g1d[7] = 0;
        const v4i z4 = {0, 0, 0, 0};
        const v8i z8i = {0, 0, 0, 0, 0, 0, 0, 0};
        __builtin_amdgcn_tensor_load_to_lds(g0d, g1d, z4, z4, z8i, 0);
        __builtin_amdgcn_s_wait_tensorcnt(0);
    }
#endif
    __syncthreads();

    const v8f z8 = {0.f,0.f,0.f,0.f,0.f,0.f,0.f,0.f};
    v8f c[2] = { z8, z8 };

    // ---- stage 1: K = 512 in steps of 4; A from LDS, B streamed from L2 ----
    for (int k4 = 0; k4 < HD / 4; ++k4) {
        const int kA   = 4 * k4 + 2 * half;
        const int kpad = kA + ((kA >> 8) << 1);   // skip 2-DWORD pad at 256
        v2f a;
        a.x = Tlds[l15 * T_STRIDE + kpad];
        a.y = Tlds[l15 * T_STRIDE + kpad + 1];
        #pragma unroll
        for (int t = 0; t < 2; ++t) {
            const int n = (2 * wave + t) * 16 + l15;
            // W1[k][n] = Wsrc[k & 127][(k >> 7)*128 + n]
            v2f b;
            b.x = Wsrc[(size_t)(kA & 127) * HD + ((kA >> 7) * D) + n];
            b.y = Wsrc[(size_t)((kA + 1) & 127) * HD + (((kA + 1) >> 7) * D) + n];
            c[t] = __builtin_amdgcn_wmma_f32_16x16x4_f32(
                       false, a, false, b, (short)0, c[t], false, false);
        }
    }
    // spill 16x128 H1 tile to LDS (C layout: VGPR r -> M = r + 8*half, N = nt*16 + l15)
    #pragma unroll
    for (int t = 0; t < 2; ++t) {
        const int n = (2 * wave + t) * 16 + l15;
        #pragma unroll
        for (int r = 0; r < 8; ++r)
            h1[(r + 8 * half) * LDS_STRIDE + n] = c[t][r];
    }
    __syncthreads();

    // ---- stage 2: K = 128 in steps of 4 ----
    v8f dacc[2] = { z8, z8 };
    for (int k4 = 0; k4 < D / 4; ++k4) {
        const int kA = 4 * k4 + 2 * half;
        v2f a;
        a.x = h1[l15 * LDS_STRIDE + kA];
        a.y = h1[l15 * LDS_STRIDE + kA + 1];
        #pragma unroll
        for (int t = 0; t < 2; ++t) {
            const int n = (2 * wave + t) * 16 + l15;
            v2f b;
            b.x = Whigh[(size_t)kA * D + n];
            b.y = Whigh[(size_t)(kA + 1) * D + n];
            dacc[t] = __builtin_amdgcn_wmma_f32_16x16x4_f32(
                          false, a, false, b, (short)0, dacc[t], false, false);
        }
    }
    #pragma unroll
    for (int t = 0; t < 2; ++t) {
        const int n = (2 * wave + t) * 16 + l15;
        const float bv = bias[n];
        #pragma unroll
        for (int r = 0; r < 8; ++r) {
            const int m = r + 8 * half;
            out[(size_t)(g0 + m) * D + n] = dacc[t][r] + bv;
        }
    }
}

// ---------------- host-side launch ----------------
extern "C" void kernel_launch(void* const* d_in, const int* in_sizes, int n_in,
                              void* d_out, int out_size, void* d_ws, size_t ws_size,
                              hipStream_t stream) {
    (void)in_sizes; (void)n_in; (void)out_size; (void)ws_size;

    const float* nf      = (const float*)d_in[0];
    const float* Wsrc    = (const float*)d_in[1];
    const float* Wdst    = (const float*)d_in[2];
    const float* att_src = (const float*)d_in[3];
    const float* att_dst = (const float*)d_in[4];
    const float* Whigh   = (const float*)d_in[5];
    const float* bhigh   = (const float*)d_in[6];
    const int*   src     = (const int*)d_in[7];
    const int*   dst     = (const int*)d_in[8];

    float* ws    = (float*)d_ws;
    float* seg   = ws + OFF_SEG;
    float* cnt   = ws + OFF_CNT;
    float* asum  = ws + OFF_ASUM;
    float* tmp   = ws + OFF_TMP;
    float* snode = ws + OFF_SNODE;
    float* shigh = ws + OFF_SHIGH;
    float* alpha = ws + OFF_ALPHA;
    float* avn   = ws + OFF_AVN;
    float* avh   = ws + OFF_AVH;
    float* amax  = ws + OFF_AMAX;

    float* out_h     = (float*)d_out;              // [N_HIGH, D]
    float* out_alpha = out_h + (size_t)N_HIGH * D; // [N_EDGES, HEADS]

    // zero accumulation buffers; set segment-max to -inf
    (void)hipMemsetAsync(ws, 0, ZERO_FLTS * sizeof(float), stream);
    k_init_amax<<<(N_HIGH * HEADS + 255) / 256, 256, 0, stream>>>((unsigned int*)amax);

    k_avecs<<<(2 * D * HEADS + 255) / 256, 256, 0, stream>>>(Wsrc, Wdst, att_src, att_dst, avn, avh);
    k_scatter_seg<<<N_EDGES / 8, 256, 0, stream>>>(nf, src, dst, seg, cnt);
    k_snode<<<N_NODES / 8, 256, 0, stream>>>(nf, avn, snode);
    k_shigh<<<N_HIGH / 8, 256, 0, stream>>>(seg, cnt, avh, shigh);
    k_alpha_max<<<(N_EDGES + 255) / 256, 256, 0, stream>>>(src, dst, snode, shigh, alpha, amax);
    k_alpha_exp<<<(N_EDGES * HEADS + 255) / 256, 256, 0, stream>>>(dst, alpha, amax, asum);
    k_norm_scatter<<<N_EDGES / 8, 256, 0, stream>>>(nf, src, dst, alpha, asum, out_alpha, tmp);
    k_gemm<<<N_HIGH / 16, 128, 0, stream>>>(tmp, Wsrc, Whigh, bhigh, out_h);
}